// MMoEB_72713796321851
// MI455X (gfx1250) — compile-verified
//
#include <hip/hip_runtime.h>
#include <math.h>

// ---------------------------------------------------------------------------
// Problem constants
// ---------------------------------------------------------------------------
#define B_SZ      64
#define HW        121        // 11*11 == D_SPECTRAL
#define D_MODEL   128
#define D_INNER   256
#define D_STATE   16
#define DT_RANK   8
#define KDIR      2
#define NROWS     (B_SZ * HW)            // 7744 (multiple of 16)
#define XDBL_M    (KDIR * (DT_RANK + 2 * D_STATE))  // 80

typedef float v2f __attribute__((ext_vector_type(2)));
typedef float v8f __attribute__((ext_vector_type(8)));

__device__ __forceinline__ v8f wmma4(v2f a, v2f b, v8f c) {
  // D = A(16x4 f32) * B(4x16 f32) + C(16x16 f32)
  return __builtin_amdgcn_wmma_f32_16x16x4_f32(false, a, false, b,
                                               (short)0, c, false, false);
}

// ---------------------------------------------------------------------------
// K1: xz[7744,512] = x[7744,128] @ in_proj_w[512,128]^T     (WMMA f32)
// one wave per 16x16 output tile; tiles = 484 * 32
// ---------------------------------------------------------------------------
__global__ __launch_bounds__(256) void k_in_proj(const float* __restrict__ x,
                                                 const float* __restrict__ W,
                                                 float* __restrict__ xz) {
  const int lane = threadIdx.x & 31;
  const int wave = threadIdx.x >> 5;
  const int tile = blockIdx.x * 8 + wave;
  const int tm = tile / 32, tn = tile % 32;
  const int row  = tm * 16 + (lane & 15);
  const int col  = tn * 16 + (lane & 15);
  const int koff = (lane >> 4) * 2;
  const float* arow = x + row * D_MODEL;
  const float* brow = W + col * D_MODEL;
  v8f c = {};
  #pragma unroll 4
  for (int k0 = 0; k0 < D_MODEL; k0 += 4) {
    v2f a = *(const v2f*)(arow + k0 + koff);
    v2f b = *(const v2f*)(brow + k0 + koff);
    c = wmma4(a, b, c);
  }
  const int cm = tm * 16 + (lane >> 4) * 8;
  const int cn = tn * 16 + (lane & 15);
  #pragma unroll
  for (int r = 0; r < 8; ++r) xz[(cm + r) * 512 + cn] = c[r];
}

// ---------------------------------------------------------------------------
// K2: conv1d(k=3, pad=1, full 256->256) over spatial axis, + bias + SiLU.
// out[b,p,o] = silu( sum_{t,i} xz[b*121+p+t-1, i] * cw[o,i,t] + cb[o] )
// 3 shifted WMMA GEMMs (one per tap), boundary rows masked branchlessly.
// tiles = 484 * 16
// ---------------------------------------------------------------------------
__global__ __launch_bounds__(256) void k_conv(const float* __restrict__ xz,
                                              const float* __restrict__ cw,
                                              const float* __restrict__ cb,
                                              float* __restrict__ xs) {
  const int lane = threadIdx.x & 31;
  const int wave = threadIdx.x >> 5;
  const int tile = blockIdx.x * 8 + wave;
  const int tm = tile / 16, tn = tile % 16;
  const int gr = tm * 16 + (lane & 15);     // row in [0, 7744)
  const int b  = gr / HW, p = gr % HW;
  const int n  = tn * 16 + (lane & 15);     // output channel
  const int koff = (lane >> 4) * 2;
  v8f c = {};
  #pragma unroll
  for (int t = 0; t < 3; ++t) {
    int ps = p + t - 1;
    float m = (ps >= 0 && ps < HW) ? 1.0f : 0.0f;
    int psc = ps < 0 ? 0 : (ps > HW - 1 ? HW - 1 : ps);
    const float* arow = xz + (b * HW + psc) * 512;   // xc half: cols [0,256)
    for (int k0 = 0; k0 < D_INNER; k0 += 4) {
      int kk = k0 + koff;
      v2f a = *(const v2f*)(arow + kk);
      a.x *= m; a.y *= m;
      v2f bf;
      bf.x = cw[n * (D_INNER * 3) + kk * 3 + t];
      bf.y = cw[n * (D_INNER * 3) + (kk + 1) * 3 + t];
      c = wmma4(a, bf, c);
    }
  }
  const int cn = tn * 16 + (lane & 15);
  const float bias = cb[cn];
  const int cm = tm * 16 + (lane >> 4) * 8;
  #pragma unroll
  for (int r = 0; r < 8; ++r) {
    float v = c[r] + bias;
    v = v / (1.0f + __expf(-v));              // SiLU
    xs[(cm + r) * D_INNER + cn] = v;
  }
}

// ---------------------------------------------------------------------------
// K3: x_dbl[b, m=80, l=256] = sum_d x_proj_w[m, d] * xs[b, d, l]
// per-b GEMM (80x121)(121x256); K=121 tail masked branchlessly.
// tiles = 64 * 5 * 16
// ---------------------------------------------------------------------------
__global__ __launch_bounds__(256) void k_xproj(const float* __restrict__ xs,
                                               const float* __restrict__ W,
                                               float* __restrict__ xdbl) {
  const int lane = threadIdx.x & 31;
  const int wave = threadIdx.x >> 5;
  const int tile = blockIdx.x * 8 + wave;
  const int b   = tile / 80;
  const int rem = tile % 80;
  const int mt = rem / 16, nt = rem % 16;
  const int m = mt * 16 + (lane & 15);      // row in [0,80)
  const int n = nt * 16 + (lane & 15);
  const int koff = (lane >> 4) * 2;
  const float* arow  = W + m * HW;
  const float* bbase = xs + b * HW * D_INNER;
  v8f c = {};
  for (int k0 = 0; k0 < 124; k0 += 4) {
    int kk = k0 + koff;
    int k0c = kk     < HW ? kk     : HW - 1; float m0 = kk     < HW ? 1.f : 0.f;
    int k1c = kk + 1 < HW ? kk + 1 : HW - 1; float m1 = kk + 1 < HW ? 1.f : 0.f;
    v2f a, bf;
    a.x  = arow[k0c] * m0;
    a.y  = arow[k1c] * m1;
    bf.x = bbase[k0c * D_INNER + n] * m0;
    bf.y = bbase[k1c * D_INNER + n] * m1;
    c = wmma4(a, bf, c);
  }
  const int cm = mt * 16 + (lane >> 4) * 8;
  const int cn = nt * 16 + (lane & 15);
  #pragma unroll
  for (int r = 0; r < 8; ++r)
    xdbl[b * XDBL_M * D_INNER + (cm + r) * D_INNER + cn] = c[r];
}

// ---------------------------------------------------------------------------
// K4: delta[b,k,d,l] = softplus( sum_r x_dbl[b, k*40+r, l]*dtw[k,d,r] + dtb[k,d] )
// ---------------------------------------------------------------------------
__global__ __launch_bounds__(256) void k_delta(const float* __restrict__ xdbl,
                                               const float* __restrict__ dtw,
                                               const float* __restrict__ dtb,
                                               float* __restrict__ delta) {
  int idx = blockIdx.x * blockDim.x + threadIdx.x;
  const int total = B_SZ * KDIR * HW * D_INNER;
  if (idx >= total) return;
  int l = idx & 255;
  int t = idx >> 8;
  int d = t % HW; t /= HW;
  int k = t & 1;
  int b = t >> 1;
  const float* dr = xdbl + b * XDBL_M * D_INNER + (k * 40) * D_INNER + l;
  const float* w  = dtw + (k * HW + d) * DT_RANK;
  float acc = dtb[k * HW + d];
  #pragma unroll
  for (int r = 0; r < DT_RANK; ++r) acc += dr[r * D_INNER] * w[r];
  delta[idx] = (acc > 20.0f) ? acc : log1pf(__expf(acc));   // softplus
}

// ---------------------------------------------------------------------------
// K5: selective-scan. 16-lane group per (b,d); lane = state n.
// For k=0: h-scan over l ascending, y written; for k=1: same scan, result
// accumulated at flipped column (y[:,0] + flip(y[:,1])). No atomics needed:
// the same lane owns both passes for a given (b,d).
// waves = 7744/2 -> 484 blocks
// ---------------------------------------------------------------------------
__global__ __launch_bounds__(256) void k_scan(const float* __restrict__ delta,
                                              const float* __restrict__ xs,
                                              const float* __restrict__ xdbl,
                                              const float* __restrict__ A_logs,
                                              const float* __restrict__ Ds,
                                              float* __restrict__ ycomb) {
  const int lane = threadIdx.x & 31;
  const int wave = threadIdx.x >> 5;
  const int g = lane >> 4;                  // half-wave group
  const int n = lane & 15;                  // state index
  const int tup = (blockIdx.x * 8 + wave) * 2 + g;   // (b,d), 7744 total
  const int b = tup / HW, d = tup % HW;
  const float* xsrow = xs + (b * HW + d) * D_INNER;
  float* yrow = ycomb + (b * HW + d) * D_INNER;
  for (int k = 0; k < 2; ++k) {
    const float Av = -__expf(A_logs[(k * HW + d) * D_STATE + n]);
    const float Dv = Ds[k * HW + d];
    const float* drow = delta + ((b * 2 + k) * HW + d) * D_INNER;
    const float* Brow = xdbl + b * XDBL_M * D_INNER + (k * 40 + DT_RANK + n) * D_INNER;
    const float* Crow = xdbl + b * XDBL_M * D_INNER + (k * 40 + DT_RANK + D_STATE + n) * D_INNER;
    float h = 0.0f;
    for (int l = 0; l < D_INNER; ++l) {
      float dl = drow[l];
      float xv = xsrow[l];
      float Bv = Brow[l];
      float Cv = Crow[l];
      float dA = __expf(dl * Av);
      h = dA * h + dl * Bv * xv;
      float yt = h * Cv;
      yt += __shfl_xor(yt, 1, 16);
      yt += __shfl_xor(yt, 2, 16);
      yt += __shfl_xor(yt, 4, 16);
      yt += __shfl_xor(yt, 8, 16);
      if (n == 0) {
        float yv = yt + Dv * xv;
        if (k == 0) yrow[l] = yv;
        else        yrow[255 - l] += yv;    // flip over l, same thread owns row
      }
    }
  }
}

// ---------------------------------------------------------------------------
// K6: LayerNorm over d=121 per (b,c=l), then gate with exact GELU(z).
// one wave per (b,c); 16384 waves
// ---------------------------------------------------------------------------
__global__ __launch_bounds__(256) void k_norm_gate(const float* __restrict__ ycomb,
                                                   const float* __restrict__ nw,
                                                   const float* __restrict__ nb,
                                                   const float* __restrict__ xz,
                                                   float* __restrict__ gated) {
  const int lane = threadIdx.x & 31;
  const int wave = threadIdx.x >> 5;
  const int pair = blockIdx.x * 8 + wave;   // (b, c)
  const int b = pair >> 8;
  const int c = pair & 255;
  const float* base = ycomb + b * HW * D_INNER + c;
  float v[4];
  float s = 0.f, s2 = 0.f;
  #pragma unroll
  for (int i = 0; i < 4; ++i) {
    int p = lane + i * 32;
    float x = (p < HW) ? base[p * D_INNER] : 0.f;
    v[i] = x;
    s += x; s2 += x * x;
  }
  #pragma unroll
  for (int off = 1; off < 32; off <<= 1) {
    s  += __shfl_xor(s,  off, 32);
    s2 += __shfl_xor(s2, off, 32);
  }
  const float mu  = s * (1.0f / 121.0f);
  const float var = s2 * (1.0f / 121.0f) - mu * mu;
  const float inv = rsqrtf(var + 1e-5f);
  #pragma unroll
  for (int i = 0; i < 4; ++i) {
    int p = lane + i * 32;
    if (p < HW) {
      float yn = (v[i] - mu) * inv * nw[p] + nb[p];
      float z  = xz[(b * HW + p) * 512 + D_INNER + c];
      float gl = 0.5f * z * (1.0f + erff(z * 0.70710678118654752f));  // exact GELU
      gated[(b * HW + p) * D_INNER + c] = yn * gl;
    }
  }
}

// ---------------------------------------------------------------------------
// K7: out[7744,128] = gated[7744,256] @ out_proj_w[128,256]^T   (WMMA f32)
// tiles = 484 * 8
// ---------------------------------------------------------------------------
__global__ __launch_bounds__(256) void k_out_proj(const float* __restrict__ gated,
                                                  const float* __restrict__ W,
                                                  float* __restrict__ out) {
  const int lane = threadIdx.x & 31;
  const int wave = threadIdx.x >> 5;
  const int tile = blockIdx.x * 8 + wave;
  const int tm = tile / 8, tn = tile % 8;
  const int row  = tm * 16 + (lane & 15);
  const int col  = tn * 16 + (lane & 15);
  const int koff = (lane >> 4) * 2;
  const float* arow = gated + row * D_INNER;
  const float* brow = W + col * D_INNER;
  v8f c = {};
  #pragma unroll 4
  for (int k0 = 0; k0 < D_INNER; k0 += 4) {
    v2f a = *(const v2f*)(arow + k0 + koff);
    v2f b = *(const v2f*)(brow + k0 + koff);
    c = wmma4(a, b, c);
  }
  const int cm = tm * 16 + (lane >> 4) * 8;
  const int cn = tn * 16 + (lane & 15);
  #pragma unroll
  for (int r = 0; r < 8; ++r) out[(cm + r) * D_MODEL + cn] = c[r];
}

// ---------------------------------------------------------------------------
// Host launcher
// ---------------------------------------------------------------------------
extern "C" void kernel_launch(void* const* d_in, const int* in_sizes, int n_in,
                              void* d_out, int out_size, void* d_ws, size_t ws_size,
                              hipStream_t stream) {
  const float* x     = (const float*)d_in[0];   // (64,11,11,128)
  const float* ipw   = (const float*)d_in[1];   // (512,128)
  const float* cw    = (const float*)d_in[2];   // (256,256,3)
  const float* cb    = (const float*)d_in[3];   // (256,)
  const float* xpw   = (const float*)d_in[4];   // (2,40,121)
  const float* dtw   = (const float*)d_in[5];   // (2,121,8)
  const float* dtb   = (const float*)d_in[6];   // (2,121)
  const float* Alogs = (const float*)d_in[7];   // (242,16)
  const float* Ds    = (const float*)d_in[8];   // (242,)
  const float* nw    = (const float*)d_in[9];   // (121,)
  const float* nb    = (const float*)d_in[10];  // (121,)
  const float* opw   = (const float*)d_in[11];  // (128,256)
  float* out = (float*)d_out;                   // (64,11,11,128)

  // workspace layout (floats)
  float* ws = (float*)d_ws;
  float* xz    = ws;                               // 7744*512  = 3,964,928
  float* xs    = xz    + (size_t)NROWS * 512;      // 7744*256  = 1,982,464
  float* xdbl  = xs    + (size_t)NROWS * D_INNER;  // 64*80*256 = 1,310,720
  float* delta = xdbl  + (size_t)B_SZ * XDBL_M * D_INNER;  // 3,964,928
  float* ycomb = delta + (size_t)B_SZ * KDIR * HW * D_INNER; // 1,982,464
  float* gated = ycomb + (size_t)NROWS * D_INNER;  // 1,982,464
  (void)in_sizes; (void)n_in; (void)out_size; (void)ws_size;

  // K1: in_proj GEMM, 484*32 tiles, 8 waves/block
  k_in_proj<<<(484 * 32) / 8, 256, 0, stream>>>(x, ipw, xz);
  // K2: conv + bias + SiLU, 484*16 tiles
  k_conv<<<(484 * 16) / 8, 256, 0, stream>>>(xz, cw, cb, xs);
  // K3: x_proj per-batch GEMM, 64*5*16 tiles
  k_xproj<<<(64 * 5 * 16) / 8, 256, 0, stream>>>(xs, xpw, xdbl);
  // K4: delta (rank-8 proj + softplus)
  {
    const int total = B_SZ * KDIR * HW * D_INNER;
    k_delta<<<(total + 255) / 256, 256, 0, stream>>>(xdbl, dtw, dtb, delta);
  }
  // K5: selective scan + direction combine, 7744 (b,d) tuples / 2 per wave
  k_scan<<<(NROWS / 2) / 8, 256, 0, stream>>>(delta, xs, xdbl, Alogs, Ds, ycomb);
  // K6: layernorm + GELU gate, 64*256 waves
  k_norm_gate<<<(B_SZ * D_INNER) / 8, 256, 0, stream>>>(ycomb, nw, nb, xz, gated);
  // K7: out_proj GEMM, 484*8 tiles
  k_out_proj<<<(484 * 8) / 8, 256, 0, stream>>>(gated, opw, out);
}